// plaCell_38302518346138
// MI455X (gfx1250) — compile-verified
//
#include <hip/hip_runtime.h>
#include <hip/hip_bf16.h>
#include <math.h>

#define B_SZ 64
#define D_SZ 1024
#define H_SZ 1024

typedef __attribute__((ext_vector_type(2))) float v2f;
typedef __attribute__((ext_vector_type(4))) float f4;
typedef __attribute__((ext_vector_type(8))) float v8f;

struct GemmPtrs { const float* W[6]; };
struct BiasPtrs { const float* Bias[6]; };

// ---------------------------------------------------------------------------
// Kernel 1: LayerNorm over D per batch row.  64 blocks x 256 threads, 4 elem/thread.
// ---------------------------------------------------------------------------
__global__ void ln_kernel(const float* __restrict__ x,
                          const float* __restrict__ gamma,
                          const float* __restrict__ beta,
                          float* __restrict__ xn) {
    int b = blockIdx.x;
    int tid = threadIdx.x;                       // 0..255
    const f4* xr = (const f4*)(x + (size_t)b * D_SZ);
    f4 v = xr[tid];
    float s  = v.x + v.y + v.z + v.w;
    float s2 = v.x * v.x + v.y * v.y + v.z * v.z + v.w * v.w;

    __shared__ float ls[256];
    __shared__ float ls2[256];
    ls[tid] = s; ls2[tid] = s2;
    __syncthreads();
    for (int off = 128; off > 0; off >>= 1) {
        if (tid < off) { ls[tid] += ls[tid + off]; ls2[tid] += ls2[tid + off]; }
        __syncthreads();
    }
    float mu  = ls[0] * (1.0f / D_SZ);
    float var = ls2[0] * (1.0f / D_SZ) - mu * mu;
    float inv = rsqrtf(var + 1e-5f);

    const f4* g4 = (const f4*)gamma;
    const f4* b4 = (const f4*)beta;
    f4 g = g4[tid], bb = b4[tid], o;
    o.x = (v.x - mu) * inv * g.x + bb.x;
    o.y = (v.y - mu) * inv * g.y + bb.y;
    o.z = (v.z - mu) * inv * g.z + bb.z;
    o.w = (v.w - mu) * inv * g.w + bb.w;
    ((f4*)(xn + (size_t)b * D_SZ))[tid] = o;
}

// ---------------------------------------------------------------------------
// Kernel 2: six projections  raw6[j][b][h] = xn @ W_j  via V_WMMA_F32_16X16X4_F32.
// One wave (32 threads) per 16x16 output tile; K-loop of 256 WMMAs.
// grid.x = 6 * (H/16) = 384 N-tiles, grid.y = B/16 = 4 M-tiles.
// A layout: lane l (l<16): A[m0+l][k+0..1]; lanes 16-31: A[m0+l-16][k+2..3]
// B layout: lane l (l<16): B[k+0..1][n0+l]; lanes 16-31: B[k+2..3][n0+l-16]
// ---------------------------------------------------------------------------
__global__ void gemm_wmma_kernel(const float* __restrict__ xn, GemmPtrs wp,
                                 float* __restrict__ raw6) {
    int tx = blockIdx.x, ty = blockIdx.y;
    int j  = tx >> 6;                 // which weight matrix (64 N-tiles per matrix)
    int n0 = (tx & 63) << 4;
    int m0 = ty << 4;
    const float* W = wp.W[j];

    int lane = threadIdx.x & 31;
    int half = lane >> 4;             // 0 or 1
    int r    = lane & 15;

    const float* arow = xn + (size_t)(m0 + r) * D_SZ + 2 * half;      // + k0 -> A[m][k0+2h]
    const float* bcol = W + (size_t)(2 * half) * H_SZ + n0 + r;       // + k0*H -> B[k0+2h][n]

    v8f acc = {};
    for (int k0 = 0; k0 < D_SZ; k0 += 4) {
        v2f a = *(const v2f*)(arow + k0);          // A[m][k0+2h], A[m][k0+2h+1]
        v2f bfrag;
        bfrag.x = bcol[(size_t)k0 * H_SZ];         // B[k0+2h][n]
        bfrag.y = bcol[(size_t)(k0 + 1) * H_SZ];   // B[k0+2h+1][n]
        acc = __builtin_amdgcn_wmma_f32_16x16x4_f32(
            /*neg_a=*/false, a, /*neg_b=*/false, bfrag,
            /*c_mod=*/(short)0, acc, /*reuse_a=*/false, /*reuse_b=*/false);
    }

    // D layout: VGPR v, lane l -> row (v + 8*half), col r
    float* out = raw6 + ((size_t)j * B_SZ + m0) * H_SZ + n0;
#pragma unroll
    for (int vr = 0; vr < 8; ++vr) {
        int m = vr + 8 * half;
        out[(size_t)m * H_SZ + r] = acc[vr];
    }
}

// ---------------------------------------------------------------------------
// Kernel 3: gate nonlinearities + n_t, m_t, divisor.  One block per batch (256 thr).
// ---------------------------------------------------------------------------
__global__ void gates_kernel(const float* __restrict__ raw6, BiasPtrs bp,
                             const float* __restrict__ nprev,
                             const float* __restrict__ mprev,
                             float* __restrict__ g_f, float* __restrict__ g_ip,
                             float* __restrict__ g_k, float* __restrict__ g_v,
                             float* __restrict__ g_q, float* __restrict__ g_o,
                             float* __restrict__ divisor,
                             float* __restrict__ n_out, float* __restrict__ m_out) {
    int b = blockIdx.x;
    int tid = threadIdx.x;
    __shared__ float red[256];
    const float inv_sqrt_h = 0.03125f;  // 1/sqrt(1024)
    float part = 0.0f;

#pragma unroll
    for (int e = 0; e < 4; ++e) {
        int h = tid * 4 + e;
        size_t idx = (size_t)b * H_SZ + h;
        size_t base = (size_t)b * H_SZ + h;
        const size_t mstride = (size_t)B_SZ * H_SZ;

        float ri = raw6[0 * mstride + base] + bp.Bias[0][h];
        float rf = raw6[1 * mstride + base] + bp.Bias[1][h];
        float ro = raw6[2 * mstride + base] + bp.Bias[2][h];
        float rq = raw6[3 * mstride + base] + bp.Bias[3][h];
        float rk = raw6[4 * mstride + base] * inv_sqrt_h + bp.Bias[4][h];
        float rv = raw6[5 * mstride + base] + bp.Bias[5][h];

        ri = fminf(fmaxf(ri, -100.0f), 100.0f);
        rf = fminf(fmaxf(rf, -100.0f), 100.0f);
        ro = fminf(fmaxf(ro, -100.0f), 100.0f);

        float i_t = expf(ri);
        float f_t = 1.0f / (1.0f + expf(-rf));
        float o_t = 1.0f / (1.0f + expf(-ro));

        float m_t = fmaxf(logf(fmaxf(f_t, 1e-8f)) + mprev[idx],
                          logf(fmaxf(i_t, 1e-8f)));
        float ip  = expf(fminf(i_t - m_t, 20.0f));   // faithful: uses i_t, not i_tilda
        float n_t = f_t * nprev[idx] + ip * rk;
        part += n_t * rq;

        g_f[idx] = f_t; g_ip[idx] = ip; g_k[idx] = rk;
        g_v[idx] = rv;  g_q[idx] = rq;  g_o[idx] = o_t;
        n_out[idx] = n_t; m_out[idx] = m_t;
    }

    red[tid] = part;
    __syncthreads();
    for (int off = 128; off > 0; off >>= 1) {
        if (tid < off) red[tid] += red[tid + off];
        __syncthreads();
    }
    if (tid == 0) divisor[b] = fmaxf(fabsf(red[0]), 1.0f);
}

// ---------------------------------------------------------------------------
// Kernel 4: streaming matrix-memory update (the 512 MB bandwidth kernel).
//   C_t[b,i,k] = f[b,i]*C[b,i,k] + (i'[b,i]*v[b,i])*k_t[b,k]
//   h_t[b,i]   = o[b,i] * (sum_k C_t[b,i,k]*q[b,k]) / divisor[b]
// Block = 256 threads = 8 waves; each wave owns one row i; k/q staged in LDS.
// Nontemporal b128 loads/stores: C (256MB) exceeds L2, streamed once.
// ---------------------------------------------------------------------------
__global__ void cupdate_kernel(const float* __restrict__ C_in,
                               const float* __restrict__ g_f,
                               const float* __restrict__ g_ip,
                               const float* __restrict__ g_k,
                               const float* __restrict__ g_v,
                               const float* __restrict__ g_q,
                               const float* __restrict__ g_o,
                               const float* __restrict__ divisor,
                               float* __restrict__ C_out,
                               float* __restrict__ h_out) {
    int b    = blockIdx.y;
    int lane = threadIdx.x & 31;
    int wave = threadIdx.x >> 5;                 // 0..7

    __shared__ float kq[2 * H_SZ];               // 8 KB: k row then q row
    {
        const f4* ks = (const f4*)(g_k + (size_t)b * H_SZ);
        const f4* qs = (const f4*)(g_q + (size_t)b * H_SZ);
        ((f4*)kq)[threadIdx.x]          = ks[threadIdx.x];
        ((f4*)(kq + H_SZ))[threadIdx.x] = qs[threadIdx.x];
    }
    __syncthreads();

    int i = blockIdx.x * 8 + wave;
    size_t ridx = (size_t)b * H_SZ + i;
    float f   = g_f[ridx];
    float ipv = g_ip[ridx] * g_v[ridx];

    const f4* crow = (const f4*)(C_in  + ((size_t)b * H_SZ + i) * H_SZ);
    f4*       orow = (f4*)      (C_out + ((size_t)b * H_SZ + i) * H_SZ);
    const f4* kl = (const f4*)kq;
    const f4* ql = (const f4*)(kq + H_SZ);

    float acc = 0.0f;
#pragma unroll
    for (int c = 0; c < 8; ++c) {
        int idx4 = c * 32 + lane;                // coalesced 512B per wave per step
        f4 cv = __builtin_nontemporal_load(&crow[idx4]);
        f4 kv = kl[idx4];
        f4 qv = ql[idx4];
        f4 cn;
        cn.x = f * cv.x + ipv * kv.x;
        cn.y = f * cv.y + ipv * kv.y;
        cn.z = f * cv.z + ipv * kv.z;
        cn.w = f * cv.w + ipv * kv.w;
        acc += cn.x * qv.x + cn.y * qv.y + cn.z * qv.z + cn.w * qv.w;
        __builtin_nontemporal_store(cn, &orow[idx4]);
    }

    // wave32 reduction
#pragma unroll
    for (int off = 16; off > 0; off >>= 1)
        acc += __shfl_xor(acc, off, 32);
    if (lane == 0)
        h_out[ridx] = g_o[ridx] * acc / divisor[b];
}

// ---------------------------------------------------------------------------
extern "C" void kernel_launch(void* const* d_in, const int* in_sizes, int n_in,
                              void* d_out, int out_size, void* d_ws, size_t ws_size,
                              hipStream_t stream) {
    const float* x     = (const float*)d_in[0];
    const float* C     = (const float*)d_in[1];
    const float* nprev = (const float*)d_in[2];
    const float* mprev = (const float*)d_in[3];
    GemmPtrs wp;
    BiasPtrs bp;
    for (int j = 0; j < 6; ++j) {
        wp.W[j]    = (const float*)d_in[4 + j];    // W_i, W_f, W_o, W_q, W_k, W_v
        bp.Bias[j] = (const float*)d_in[10 + j];   // B_i, B_f, B_o, B_q, B_k, B_v
    }
    const float* gamma = (const float*)d_in[16];
    const float* beta  = (const float*)d_in[17];

    const size_t BH  = (size_t)B_SZ * H_SZ;            // 65536
    const size_t BHH = (size_t)B_SZ * H_SZ * H_SZ;     // 67108864

    float* out   = (float*)d_out;                      // tuple: h_t, C_t, n_t, m_t
    float* h_out = out;
    float* C_out = out + BH;
    float* n_out = out + BH + BHH;
    float* m_out = n_out + BH;

    float* ws     = (float*)d_ws;
    float* xn     = ws;                 // BH
    float* raw6   = xn + BH;            // 6*BH
    float* g_f    = raw6 + 6 * BH;
    float* g_ip   = g_f  + BH;
    float* g_k    = g_ip + BH;
    float* g_v    = g_k  + BH;
    float* g_q    = g_v  + BH;
    float* g_o    = g_q  + BH;
    float* divv   = g_o  + BH;          // 64 floats

    ln_kernel<<<B_SZ, 256, 0, stream>>>(x, gamma, beta, xn);

    dim3 g2(6 * (H_SZ / 16), B_SZ / 16);               // 384 x 4 tiles, 1 wave each
    gemm_wmma_kernel<<<g2, 32, 0, stream>>>(xn, wp, raw6);

    gates_kernel<<<B_SZ, 256, 0, stream>>>(raw6, bp, nprev, mprev,
                                           g_f, g_ip, g_k, g_v, g_q, g_o,
                                           divv, n_out, m_out);

    dim3 g4(H_SZ / 8, B_SZ);                           // 8 rows per 256-thread block
    cupdate_kernel<<<g4, 256, 0, stream>>>(C, g_f, g_ip, g_k, g_v, g_q, g_o,
                                           divv, C_out, h_out);
}